// SparseGPMatern52_10814727651969
// MI455X (gfx1250) — compile-verified
//
#include <hip/hip_runtime.h>
#include <hip/hip_bf16.h>

typedef __attribute__((ext_vector_type(2))) float v2f;
typedef __attribute__((ext_vector_type(8))) float v8f;

#define M_IND 50
#define MPAD  64
#define BST   66   // padded LDS row stride (floats), 66*4 = 264 bytes -> keeps b64 alignment

// workspace layout (floats)
#define WS_AINV  0        // 64*64 padded inverse (symmetric, zero-padded)
#define WS_ZS    4096     // 64*5 scaled inducing points (rows >=50 zero)
#define WS_Z2    4416     // 64
#define WS_ILS   4480     // 5
#define WS_PV    4485     // 1

__device__ __constant__ float cZSCALE[5] = {0.25f, 0.2f, 0.08f, 400.0f, 10.0f};
__device__ __constant__ float cZSHIFT[5] = {0.0f, 0.0f, 0.0f, 800.0f, 12.0f};

// ---------------- Kernel 1: preprocessing + 50x50 Gauss-Jordan inverse ----------------
__global__ void __launch_bounds__(256)
gp_prep_kernel(const float* __restrict__ log_ls,
               const float* __restrict__ log_var,
               const float* __restrict__ Z_raw,
               float* __restrict__ ws) {
    __shared__ float Aaug[M_IND * 100];   // [50][100] augmented
    __shared__ float fcol[M_IND];
    __shared__ float Zsl[M_IND * 5];
    __shared__ float z2l[M_IND];
    __shared__ float invl[5];
    __shared__ float pvs;

    const int t = threadIdx.x;

    if (t < 5)  invl[t] = 1.0f / (__expf(log_ls[t]) + 1e-8f);
    if (t == 5) pvs = __expf(log_var[0]);
    __syncthreads();

    for (int i = t; i < M_IND * 5; i += 256) {
        const int d = i % 5;
        const float z = tanhf(Z_raw[i]) * cZSCALE[d] + cZSHIFT[d];
        Zsl[i] = z * invl[d];
    }
    __syncthreads();

    if (t < M_IND) {
        float s = 0.0f;
        #pragma unroll
        for (int d = 0; d < 5; ++d) { const float v = Zsl[t * 5 + d]; s += v * v; }
        z2l[t] = s;
    }
    __syncthreads();

    const float pv = pvs;
    for (int idx = t; idx < M_IND * M_IND; idx += 256) {
        const int i = idx / M_IND, j = idx % M_IND;
        float d2 = 0.0f;
        #pragma unroll
        for (int d = 0; d < 5; ++d) {
            const float df = Zsl[i * 5 + d] - Zsl[j * 5 + d];
            d2 += df * df;
        }
        const float dd = d2 + 1e-8f;
        const float dv = sqrtf(dd);
        const float s  = 2.2360679774997896f * dv;  // sqrt(5)*d
        float k = pv * (1.0f + s + 1.6666666666666667f * dd) * __expf(-s);
        if (i == j) k += 1e-4f;
        Aaug[i * 100 + j]          = k;
        Aaug[i * 100 + M_IND + j]  = (i == j) ? 1.0f : 0.0f;
    }
    __syncthreads();

    // Gauss-Jordan, no pivoting (SPD + jitter)
    for (int p = 0; p < M_IND; ++p) {
        const float rp = 1.0f / Aaug[p * 100 + p];
        __syncthreads();
        for (int c = t; c < 100; c += 256) Aaug[p * 100 + c] *= rp;
        __syncthreads();
        if (t < M_IND) fcol[t] = Aaug[t * 100 + p];
        __syncthreads();
        for (int idx = t; idx < M_IND * 100; idx += 256) {
            const int i = idx / 100, c = idx % 100;
            if (i != p) Aaug[i * 100 + c] -= fcol[i] * Aaug[p * 100 + c];
        }
        __syncthreads();
    }

    // write padded outputs
    for (int idx = t; idx < MPAD * MPAD; idx += 256) {
        const int i = idx >> 6, j = idx & 63;
        ws[WS_AINV + idx] = (i < M_IND && j < M_IND) ? Aaug[i * 100 + M_IND + j] : 0.0f;
    }
    for (int idx = t; idx < MPAD * 5; idx += 256)
        ws[WS_ZS + idx] = (idx < M_IND * 5) ? Zsl[idx] : 0.0f;
    for (int idx = t; idx < MPAD; idx += 256)
        ws[WS_Z2 + idx] = (idx < M_IND) ? z2l[idx] : 0.0f;
    if (t < 5)  ws[WS_ILS + t] = invl[t];
    if (t == 5) ws[WS_PV] = pvs;
}

// ---------------- Kernel 2: main — Matern evals (VALU) + quadratic form (WMMA f32) ----------------
__global__ void __launch_bounds__(256)
gp_var_kernel(const float* __restrict__ x_star,
              const float* __restrict__ ws,
              float* __restrict__ out,
              int nq) {
    // LDS: K^-1 in fragment-friendly layout + per-wave k rows + constants (~52 KB)
    __shared__ __align__(16) float Blds[MPAD * BST];   // Blds[c*BST + r] == Ainv[r][c] (symmetric)
    __shared__ __align__(16) float Klds[8 * 16 * BST]; // per-wave 16x64 k_xZ rows (padded)
    __shared__ float Zs2[MPAD * 5];
    __shared__ float z2s[MPAD];
    __shared__ float il[5];
    __shared__ float pvv;

    const int t = threadIdx.x;
    for (int idx = t; idx < MPAD * MPAD; idx += 256) {
        const int c = idx >> 6, r = idx & 63;
        Blds[c * BST + r] = ws[WS_AINV + idx];
    }
    for (int idx = t; idx < MPAD * 5; idx += 256) Zs2[idx] = ws[WS_ZS + idx];
    for (int idx = t; idx < MPAD; idx += 256)     z2s[idx] = ws[WS_Z2 + idx];
    if (t < 5)  il[t] = ws[WS_ILS + t];
    if (t == 5) pvv = ws[WS_PV];
    __syncthreads();

    const int wave = t >> 5;
    const int lane = t & 31;
    const int m  = lane & 15;   // query row within wave tile / N column
    const int hf = lane >> 4;   // half selects K-pair {0,1} vs {2,3}
    const int base = blockIdx.x * 128 + wave * 16;

    // load this lane's query (clamped; real guard at the store)
    int q = base + m;
    const int qc = (q < nq) ? q : (nq - 1);
    float xs[5];
    float x2 = 0.0f;
    #pragma unroll
    for (int d = 0; d < 5; ++d) {
        xs[d] = x_star[(size_t)qc * 5 + d] * il[d];
        x2 += xs[d] * xs[d];
    }
    const float pv = pvv;

    // A fragments: lane holds k(q, j) for j = 4*kk + 2*hf + {0,1}
    float aA[16], aB[16];
    float* Kl = &Klds[wave * 16 * BST];
    #pragma unroll
    for (int kk = 0; kk < 16; ++kk) {
        const int j0 = 4 * kk + 2 * hf;
        #pragma unroll
        for (int u = 0; u < 2; ++u) {
            const int j = j0 + u;
            float kv = 0.0f;
            if (j < M_IND) {
                float dot = 0.0f;
                #pragma unroll
                for (int d = 0; d < 5; ++d) dot += xs[d] * Zs2[j * 5 + d];
                float d2 = x2 + z2s[j] - 2.0f * dot;
                d2 = fmaxf(d2, 0.0f) + 1e-8f;
                const float dv = sqrtf(d2);
                const float s  = 2.2360679774997896f * dv;
                kv = pv * (1.0f + s + 1.6666666666666667f * d2) * __expf(-s);
            }
            if (u == 0) aA[kk] = kv; else aB[kk] = kv;
            Kl[m * BST + j0 + u] = kv;   // same-wave LDS, in-order: no barrier needed
        }
    }

    // GEMM T = Kq(16x64) @ Ainv(64x64) on the matrix pipe: 64 x v_wmma_f32_16x16x4_f32
    v8f cacc[4];
    #pragma unroll
    for (int n = 0; n < 4; ++n)
        #pragma unroll
        for (int r = 0; r < 8; ++r) cacc[n][r] = 0.0f;

    #pragma unroll
    for (int kk = 0; kk < 16; ++kk) {
        v2f a; a[0] = aA[kk]; a[1] = aB[kk];
        #pragma unroll
        for (int n = 0; n < 4; ++n) {
            const int boff = (16 * n + m) * BST + 4 * kk + 2 * hf;
            v2f b; b[0] = Blds[boff]; b[1] = Blds[boff + 1];
            cacc[n] = __builtin_amdgcn_wmma_f32_16x16x4_f32(
                false, a, false, b, (short)0, cacc[n], false, false);
        }
    }

    // red[row] = sum_col T[row][col] * k[row][col]
    // lane (hf,m) holds T[r+8*hf][16n+m] in cacc[n][r]; k rows come from this wave's LDS tile
    float contrib[8];
    #pragma unroll
    for (int r = 0; r < 8; ++r) {
        float s = 0.0f;
        #pragma unroll
        for (int n = 0; n < 4; ++n)
            s += cacc[n][r] * Kl[(r + 8 * hf) * BST + 16 * n + m];
        contrib[r] = s;
    }

    #pragma unroll
    for (int r = 0; r < 8; ++r) {
        float v = contrib[r];
        v += __shfl_xor(v, 1, 32);
        v += __shfl_xor(v, 2, 32);
        v += __shfl_xor(v, 4, 32);
        v += __shfl_xor(v, 8, 32);   // full sum within each 16-lane half
        if (m == r) {
            const int qo = base + r + 8 * hf;
            if (qo < nq) out[qo] = sqrtf(fmaxf(pv - v, 1e-6f));
        }
    }
}

extern "C" void kernel_launch(void* const* d_in, const int* in_sizes, int n_in,
                              void* d_out, int out_size, void* d_ws, size_t ws_size,
                              hipStream_t stream) {
    const float* x_star  = (const float*)d_in[0];
    const float* log_ls  = (const float*)d_in[1];
    const float* log_var = (const float*)d_in[2];
    const float* Z_raw   = (const float*)d_in[3];
    float* outp = (float*)d_out;
    float* ws   = (float*)d_ws;

    const int nq = in_sizes[0] / 5;

    gp_prep_kernel<<<1, 256, 0, stream>>>(log_ls, log_var, Z_raw, ws);

    const int blocks = (nq + 127) / 128;
    gp_var_kernel<<<blocks, 256, 0, stream>>>(x_star, ws, outp, nq);
}